// GNN_layer_66752381714632
// MI455X (gfx1250) — compile-verified
//
#include <hip/hip_runtime.h>
#include <hip/hip_bf16.h>

#define HID    128
#define BN_EPS 1e-5f

typedef __attribute__((ext_vector_type(2))) float v2f;
typedef __attribute__((ext_vector_type(8))) float v8f;

// ---------------------------------------------------------------------------
// fp32 WMMA: D(16x16) = A(16x4) * B(4x16) + C   (V_WMMA_F32_16X16X4_F32)
// ---------------------------------------------------------------------------
__device__ __forceinline__ v8f wmma_f32_16x16x4(v2f a, v2f b, v8f c) {
    return __builtin_amdgcn_wmma_f32_16x16x4_f32(
        /*neg_a=*/false, a, /*neg_b=*/false, b,
        /*c_mod=*/(short)0, c, /*reuse_a=*/false, /*reuse_b=*/false);
}

// ① agg[i][c] = b[c];  deg[i] = 1 (self loop)
__global__ void k_init(const float* __restrict__ b, float* __restrict__ agg,
                       float* __restrict__ deg, int n_elem) {
    int idx = blockIdx.x * blockDim.x + threadIdx.x;
    if (idx >= n_elem) return;
    int c = idx & (HID - 1);
    agg[idx] = b[c];
    if (c == 0) deg[idx >> 7] = 1.0f;
}

// ② deg[dst[e]] += 1
__global__ void k_degcount(const int* __restrict__ dst, float* __restrict__ deg, int E) {
    int e = blockIdx.x * blockDim.x + threadIdx.x;
    if (e < E) atomicAdd(&deg[dst[e]], 1.0f);
}

// ③ deg -> rsqrt(deg) in place
__global__ void k_rsqrt(float* __restrict__ deg, int N) {
    int i = blockIdx.x * blockDim.x + threadIdx.x;
    if (i < N) deg[i] = rsqrtf(deg[i]);
}

// ④ bias2[n] = sum_k (beta[k] - mean[k]*scale[k]) * W[k][n]   (one block, 128 thr)
__global__ void k_bias2(const float* __restrict__ gamma, const float* __restrict__ beta,
                        const float* __restrict__ mean,  const float* __restrict__ var,
                        const float* __restrict__ W, float* __restrict__ bias2) {
    int n = threadIdx.x;
    float s = 0.0f;
    for (int k = 0; k < HID; ++k) {
        float sc = rsqrtf(var[k] + BN_EPS) * gamma[k];
        float sh = beta[k] - mean[k] * sc;
        s += sh * W[k * HID + n];
    }
    bias2[n] = s;
}

// ⑤ m = x @ (diag(scale) * W) + bias2   via fp32 WMMA.
//    WG = 256 thr = 8 waves; each wave: 16 rows x 128 cols; block: 128 rows.
//    LDS holds W' pre-swizzled into B-fragment order: [ks][tile][lane][j],
//    so each lane's B fragment is one conflict-free ds_load_b64.
__global__ __launch_bounds__(256) void k_bn_gemm(
    const float* __restrict__ x, const float* __restrict__ gamma,
    const float* __restrict__ var, const float* __restrict__ W,
    const float* __restrict__ bias2, float* __restrict__ m, int N)
{
    __shared__ float sW[HID * HID];   // exactly 64 KB

    const int tid = threadIdx.x;

    // Stage W' = scale[k] * W[k][n] swizzled into fragment order.
    for (int idx = tid; idx < HID * HID; idx += 256) {
        int j    = idx & 1;
        int lane = (idx >> 1) & 31;
        int tile = (idx >> 6) & 7;
        int ks   = idx >> 9;
        int k    = 4 * ks + 2 * (lane >> 4) + j;     // B layout: K = j + 2*(lane/16)
        int n    = tile * 16 + (lane & 15);          // N = lane%16
        float sc = rsqrtf(var[k] + BN_EPS) * gamma[k];
        sW[idx]  = sc * W[k * HID + n];
    }
    __syncthreads();

    const int wave  = tid >> 5;
    const int lane  = tid & 31;
    const int khalf = lane >> 4;
    const int colb  = lane & 15;

    const int rowbase = blockIdx.x * 128 + wave * 16;
    const int r       = rowbase + colb;              // A layout: M = lane%16
    const int rc      = (r < N) ? r : (N - 1);       // clamp loads (EXEC stays all-1s)
    const float* xr   = x + (long)rc * HID;

    v8f acc[8] = {};                                 // 8 column tiles of 16

    for (int ks = 0; ks < 32; ++ks) {
        const int k0 = 4 * ks + 2 * khalf;
        v2f a;
        a.x = xr[k0];
        a.y = xr[k0 + 1];
        const float* wb = &sW[(ks * 8) * 64];
#pragma unroll
        for (int t = 0; t < 8; ++t) {
            v2f b = *(const v2f*)(wb + t * 64 + lane * 2);   // ds_load_b64, no conflicts
            acc[t] = wmma_f32_16x16x4(a, b, acc[t]);
        }
    }

    // bias2 per output column (8 cached scalar loads per lane)
    float bsv[8];
#pragma unroll
    for (int t = 0; t < 8; ++t) bsv[t] = bias2[t * 16 + colb];

    // D layout: element acc[t][i] = D[row = i + 8*khalf][col = t*16 + lane%16]
#pragma unroll
    for (int i = 0; i < 8; ++i) {
        int row = rowbase + i + 8 * khalf;
        if (row < N) {
            float* mr = m + (long)row * HID;
#pragma unroll
            for (int t = 0; t < 8; ++t)
                mr[t * 16 + colb] = acc[t][i] + bsv[t];
        }
    }
}

// ⑥ one wave per edge: agg[dst] += m[src] * dinv[src]*dinv[dst]
__global__ void k_scatter(const int* __restrict__ src, const int* __restrict__ dst,
                          const float* __restrict__ m, const float* __restrict__ dinv,
                          float* __restrict__ agg, int E) {
    int t = blockIdx.x * blockDim.x + threadIdx.x;
    int e = t >> 5;
    if (e >= E) return;
    int lane = t & 31;
    int s = src[e], d = dst[e];
    float nrm = dinv[s] * dinv[d];
    const float4 v = *(const float4*)(m + (long)s * HID + lane * 4);   // coalesced 512B row
    float* ap = agg + (long)d * HID + lane * 4;
    atomicAdd(ap + 0, v.x * nrm);
    atomicAdd(ap + 1, v.y * nrm);
    atomicAdd(ap + 2, v.z * nrm);
    atomicAdd(ap + 3, v.w * nrm);
}

// ⑦ out = x + relu(agg + m * dinv^2)   (self-loop term added here, no atomics)
__global__ void k_finalize(const float* __restrict__ x, const float* __restrict__ m,
                           const float* __restrict__ agg, const float* __restrict__ dinv,
                           float* __restrict__ out, int n_elem) {
    int idx = blockIdx.x * blockDim.x + threadIdx.x;
    if (idx >= n_elem) return;
    float di = dinv[idx >> 7];
    float v  = agg[idx] + m[idx] * di * di;
    out[idx] = x[idx] + fmaxf(v, 0.0f);
}

extern "C" void kernel_launch(void* const* d_in, const int* in_sizes, int n_in,
                              void* d_out, int out_size, void* d_ws, size_t ws_size,
                              hipStream_t stream) {
    const float* x     = (const float*)d_in[0];
    const int*   eidx  = (const int*)  d_in[1];
    const float* gamma = (const float*)d_in[2];
    const float* beta  = (const float*)d_in[3];
    const float* rmean = (const float*)d_in[4];
    const float* rvar  = (const float*)d_in[5];
    const float* W     = (const float*)d_in[6];
    const float* b     = (const float*)d_in[7];
    float*       out   = (float*)d_out;

    const int N = in_sizes[0] / HID;
    const int E = in_sizes[1] / 2;
    const int*   src = eidx;
    const int*   dst = eidx + E;

    // workspace carve-up: m[N*128] | agg[N*128] | deg/dinv[N] | bias2[128]
    float* m     = (float*)d_ws;
    float* agg   = m   + (size_t)N * HID;
    float* deg   = agg + (size_t)N * HID;
    float* bias2 = deg + N;

    const int n_elem = N * HID;

    k_init    <<<(n_elem + 255) / 256, 256, 0, stream>>>(b, agg, deg, n_elem);
    k_degcount<<<(E + 255) / 256,      256, 0, stream>>>(dst, deg, E);
    k_rsqrt   <<<(N + 255) / 256,      256, 0, stream>>>(deg, N);
    k_bias2   <<<1, HID,                    0, stream>>>(gamma, beta, rmean, rvar, W, bias2);
    k_bn_gemm <<<(N + 127) / 128,      256, 0, stream>>>(x, gamma, rvar, W, bias2, m, N);
    {
        long threads = (long)E * 32;
        k_scatter<<<(int)((threads + 255) / 256), 256, 0, stream>>>(src, dst, m, deg, agg, E);
    }
    k_finalize<<<(n_elem + 255) / 256, 256, 0, stream>>>(x, m, agg, deg, out, n_elem);
}